// GCE_GNN_Model_68453188764262
// MI455X (gfx1250) — compile-verified
//
#include <hip/hip_runtime.h>
#include <cstddef>

// ---------------- problem constants ----------------
#define BB 256
#define LL 64
#define SS 12
#define DD 100
#define NN 43098
#define NC 43097            // NUM_NODE - 1 (emb[1:])
#define LEAKY 0.2f
#define NEGV  -9000000000000000.0f

typedef __attribute__((ext_vector_type(2))) float v2f;
typedef __attribute__((ext_vector_type(8))) float v8f;

// ---------------- K1: h = emb[items] ----------------
__global__ void k_h_gather(const int* __restrict__ items,
                           const float* __restrict__ emb,
                           float* __restrict__ h) {
  int idx = blockIdx.x * blockDim.x + threadIdx.x;
  if (idx >= BB * LL * DD) return;
  int d  = idx % DD;
  int bl = idx / DD;
  h[idx] = emb[(size_t)items[bl] * DD + d];
}

// ---------------- K2: local attention -> h_local ----------------
// One block per batch b. h(64x100), A(4x100), alpha(64x65 padded) in LDS.
// alpha @ h done with V_WMMA_F32_16X16X4_F32 (4x7 tiles, K=64 in 16 steps).
__global__ void __launch_bounds__(256)
k_local(const int* __restrict__ adj,
        const float* __restrict__ a0, const float* __restrict__ a1,
        const float* __restrict__ a2, const float* __restrict__ a3,
        const float* __restrict__ h, float* __restrict__ h_local) {
  __shared__ float sh[LL][DD];       // 25600 B
  __shared__ float sA[4][DD];        //  1600 B
  __shared__ float sAl[LL][LL + 1];  // 16640 B, stride 65 -> WMMA A reads conflict-free
  int b = blockIdx.x, t = threadIdx.x;
  const float* hb = h + (size_t)b * LL * DD;
  for (int i = t; i < LL * DD; i += 256) sh[i / DD][i % DD] = hb[i];
  for (int i = t; i < DD; i += 256) {
    sA[0][i] = a0[i]; sA[1][i] = a1[i]; sA[2][i] = a2[i]; sA[3][i] = a3[i];
  }
  __syncthreads();
  const int* adjb = adj + (size_t)b * LL * LL;
  for (int p = t; p < LL * LL; p += 256) {
    int i = p / LL, j = p % LL;
    int a = adjb[p];
    float val = NEGV;
    if (a >= 1 && a <= 4) {            // only the adj-selected k contributes
      const float* Ak = sA[a - 1];
      float s = 0.f;
      for (int d = 0; d < DD; ++d) s += sh[i][d] * sh[j][d] * Ak[d];
      val = s > 0.f ? s : LEAKY * s;   // leaky relu on e
    }
    sAl[i][j] = val;
  }
  __syncthreads();
  if (t < LL) {                        // softmax over j, row t
    float mx = sAl[t][0];
    for (int j = 1; j < LL; ++j) mx = fmaxf(mx, sAl[t][j]);
    float sum = 0.f;
    for (int j = 0; j < LL; ++j) { float e = expf(sAl[t][j] - mx); sAl[t][j] = e; sum += e; }
    float inv = 1.f / sum;
    for (int j = 0; j < LL; ++j) sAl[t][j] *= inv;
  }
  __syncthreads();
  // h_local = alpha @ h via WMMA: M=64 (4 tiles), N=100 (7 tiles), K=64
  float* hl = h_local + (size_t)b * LL * DD;
  int waveid = t >> 5, lane = t & 31;
  int koff = (lane >> 4) * 2;
  for (int tile = waveid; tile < 4 * 7; tile += 8) {
    int m0 = (tile / 7) * 16, n0 = (tile % 7) * 16;
    int am  = m0 + (lane & 15);
    int nn  = n0 + (lane & 15);
    int ncl = nn < DD ? nn : DD - 1;   // clamp B column; stores guarded
    v8f c = {};
    for (int k = 0; k < LL; k += 4) {
      v2f a, bf;
      a.x  = sAl[am][k + koff];
      a.y  = sAl[am][k + koff + 1];
      bf.x = sh[k + koff][ncl];
      bf.y = sh[k + koff + 1][ncl];
      c = __builtin_amdgcn_wmma_f32_16x16x4_f32(false, a, false, bf,
                                                (short)0, c, false, false);
    }
    if (nn < DD) {
      int mb = m0 + ((lane >> 4) << 3);
#pragma unroll
      for (int v = 0; v < 8; ++v)
        hl[(size_t)(mb + v) * DD + nn] = c[v];
    }
  }
}

// ---------------- K3: sess = mean_l emb[seq_features]*mask ----------------
__global__ void __launch_bounds__(128)
k_sess(const int* __restrict__ seqf, const float* __restrict__ maskv,
       const float* __restrict__ emb, float* __restrict__ sess) {
  int b = blockIdx.x, t = threadIdx.x;
  __shared__ int   rows[LL];
  __shared__ float msum_sh;
  if (t < LL) rows[t] = seqf[b * LL + t];
  if (t == 0) {
    float ms = 0.f;
    for (int l = 0; l < LL; ++l) ms += maskv[b * LL + l];
    msum_sh = ms;
  }
  __syncthreads();
  if (t < DD) {
    float s = 0.f;
    for (int l = 0; l < LL; ++l)
      s += emb[(size_t)rows[l] * DD + t] * maskv[b * LL + l];
    sess[b * DD + t] = s / msum_sh;
  }
}

// ---------------- K4: global neighbor attention + combine ----------------
// One block per 4 l's of one batch: feat is 48x101 (M=48, 3 exact 16-tiles).
// al = leaky(feat @ g_w1) via WMMA f32 16x16x4; feat synthesized inline
// (sess[k]*nv[m][k], weight at k==100, zero-padded K 101->104, N 100->112).
__global__ void __launch_bounds__(128)
k_global(const int* __restrict__ items, const int* __restrict__ adj_all,
         const float* __restrict__ num_w, const float* __restrict__ emb,
         const float* __restrict__ sess,
         const float* __restrict__ g_w1, const float* __restrict__ g_w2,
         const float* __restrict__ g_w3,
         const float* __restrict__ h, const float* __restrict__ h_local,
         float* __restrict__ h_combine) {
  const int LB = 4;                  // l's per block
  const int MR = LB * SS;            // 48 feat rows
  __shared__ float s_sess[DD];
  __shared__ int   s_items[LB];
  __shared__ int   s_nbr[MR];
  __shared__ float s_wf[MR];
  __shared__ float s_nv[MR][DD + 1]; // 19392 B, stride 101 -> conflict-free A reads
  __shared__ float s_al[MR][DD + 1]; // 19392 B
  __shared__ float s_gl[MR];
  __shared__ float s_h4[LB][DD];
  __shared__ float s_nb[LB][DD];
  int bl0 = blockIdx.x * LB;         // first (b,l) flat index
  int b   = bl0 / LL;
  int t   = threadIdx.x;
  if (t < LB) s_items[t] = items[bl0 + t];
  if (t < DD) s_sess[t] = sess[b * DD + t];
  __syncthreads();
  if (t < MR) {
    int ll = t / SS, s = t % SS;
    s_nbr[t] = adj_all[(size_t)s_items[ll] * SS + s];
    s_wf[t]  = num_w[(size_t)s_items[ll] * SS + s];
  }
  for (int o = t; o < LB * DD; o += 128) {
    int ll = o / DD, d = o % DD;
    s_h4[ll][d] = h[(size_t)(bl0 + ll) * DD + d];
  }
  __syncthreads();
  for (int o = t; o < MR * DD; o += 128) {
    int r = o / DD, d = o % DD;
    s_nv[r][d] = emb[(size_t)s_nbr[r] * DD + d];
  }
  __syncthreads();
  {
    int waveid = t >> 5, lane = t & 31;
    int koff = (lane >> 4) * 2;
    const int NT = (DD + 15) / 16;   // 7
    const int MT = MR / 16;          // 3
    for (int tile = waveid; tile < MT * NT; tile += 4) {
      int m0 = (tile / NT) * 16, n0 = (tile % NT) * 16;
      int am = m0 + (lane & 15);
      int nn = n0 + (lane & 15);
      v8f c = {};
      for (int k = 0; k <= DD; k += 4) {           // 26 K-steps (0..103)
        int k0 = k + koff, k1 = k + koff + 1;
        v2f a, bf;
        a.x  = (k0 < DD) ? s_sess[k0] * s_nv[am][k0]
                         : (k0 == DD ? s_wf[am] : 0.f);
        a.y  = (k1 < DD) ? s_sess[k1] * s_nv[am][k1]
                         : (k1 == DD ? s_wf[am] : 0.f);
        bf.x = (k0 <= DD && nn < DD) ? g_w1[k0 * DD + nn] : 0.f;
        bf.y = (k1 <= DD && nn < DD) ? g_w1[k1 * DD + nn] : 0.f;
        c = __builtin_amdgcn_wmma_f32_16x16x4_f32(false, a, false, bf,
                                                  (short)0, c, false, false);
      }
      if (nn < DD) {
        int mb = m0 + ((lane >> 4) << 3);
#pragma unroll
        for (int v = 0; v < 8; ++v) {
          float x = c[v];
          s_al[mb + v][nn] = x > 0.f ? x : LEAKY * x;  // leaky relu
        }
      }
    }
  }
  __syncthreads();
  if (t < MR) {
    float acc = 0.f;
    for (int d = 0; d < DD; ++d) acc += s_al[t][d] * g_w2[d];
    s_gl[t] = acc;
  }
  __syncthreads();
  if (t < LB) {                      // softmax over S=12 per l
    float* g = &s_gl[t * SS];
    float mx = g[0];
    for (int s = 1; s < SS; ++s) mx = fmaxf(mx, g[s]);
    float sum = 0.f;
    for (int s = 0; s < SS; ++s) { float e = expf(g[s] - mx); g[s] = e; sum += e; }
    float inv = 1.f / sum;
    for (int s = 0; s < SS; ++s) g[s] *= inv;
  }
  __syncthreads();
  for (int o = t; o < LB * DD; o += 128) {
    int ll = o / DD, d = o % DD;
    float nb = 0.f;
    for (int s = 0; s < SS; ++s) nb += s_gl[ll * SS + s] * s_nv[ll * SS + s][d];
    s_nb[ll][d] = nb;
  }
  __syncthreads();
  for (int o = t; o < LB * DD; o += 128) {
    int ll = o / DD, d = o % DD;
    float acc = 0.f;
    for (int d2 = 0; d2 < DD; ++d2) acc += s_h4[ll][d2] * g_w3[d2 * DD + d];
    for (int d2 = 0; d2 < DD; ++d2) acc += s_nb[ll][d2] * g_w3[(DD + d2) * DD + d];
    float hg = acc > 0.f ? acc : 0.f;  // relu
    size_t idx = (size_t)(bl0 + ll) * DD + d;
    h_combine[idx] = h_local[idx] + hg;
  }
}

// ---------------- K5: seq_hidden (alias gather) + hs ----------------
__global__ void __launch_bounds__(128)
k_seqhid(const int* __restrict__ alias_inputs, const float* __restrict__ maskv,
         const float* __restrict__ h_combine,
         float* __restrict__ seq_hidden, float* __restrict__ hs) {
  int b = blockIdx.x, t = threadIdx.x;
  __shared__ int   al[LL];
  __shared__ float msum_sh;
  if (t < LL) al[t] = alias_inputs[b * LL + t];
  if (t == 0) {
    float ms = 0.f;
    for (int l = 0; l < LL; ++l) ms += maskv[b * LL + l];
    msum_sh = ms;
  }
  __syncthreads();
  if (t < DD) {
    float acc = 0.f;
    for (int l = 0; l < LL; ++l) {
      float v = h_combine[((size_t)b * LL + al[l]) * DD + t];
      seq_hidden[((size_t)b * LL + l) * DD + t] = v;
      acc += v * maskv[b * LL + l];
    }
    hs[b * DD + t] = acc / msum_sh;
  }
}

// ---------------- K6: tanh/GLU attention -> select ----------------
__global__ void __launch_bounds__(256)
k_select(const float* __restrict__ maskv, const float* __restrict__ pos_emb,
         const float* __restrict__ w_1, const float* __restrict__ w_2,
         const float* __restrict__ glu1_w, const float* __restrict__ glu1_b,
         const float* __restrict__ glu2_w, const float* __restrict__ glu2_b,
         const float* __restrict__ seq_hidden, const float* __restrict__ hs,
         float* __restrict__ selectv) {
  __shared__ float sh[LL * DD];   // 25600 B
  __shared__ float nh[LL * DD];   // 25600 B
  __shared__ float s_hs[DD];
  __shared__ float s_hsg[DD];     // hs@glu2_w + glu2_b
  __shared__ float s_beta[LL];
  int b = blockIdx.x, t = threadIdx.x;
  const float* shg = seq_hidden + (size_t)b * LL * DD;
  for (int i = t; i < LL * DD; i += 256) sh[i] = shg[i];
  if (t < DD) s_hs[t] = hs[b * DD + t];
  if (t < LL) s_beta[t] = 0.f;
  __syncthreads();
  if (t < DD) {
    float acc = glu2_b[t];
    for (int d2 = 0; d2 < DD; ++d2) acc += s_hs[d2] * glu2_w[d2 * DD + t];
    s_hsg[t] = acc;
  }
  // nh = tanh([pe, seq_hidden] @ w_1)
  for (int o = t; o < LL * DD; o += 256) {
    int l = o / DD, dout = o % DD;
    float acc = 0.f;
    for (int d2 = 0; d2 < DD; ++d2) acc += pos_emb[l * DD + d2] * w_1[d2 * DD + dout];
    for (int d2 = 0; d2 < DD; ++d2) acc += sh[l * DD + d2] * w_1[(DD + d2) * DD + dout];
    nh[o] = tanhf(acc);
  }
  __syncthreads();
  // beta_l = sum_dout sigmoid(nh@glu1_w + glu1_b + hsg) * w_2[dout]
  for (int o = t; o < LL * DD; o += 256) {
    int l = o / DD, dout = o % DD;
    float acc = glu1_b[dout] + s_hsg[dout];
    for (int d2 = 0; d2 < DD; ++d2) acc += nh[l * DD + d2] * glu1_w[d2 * DD + dout];
    float sig = 1.f / (1.f + expf(-acc));
    atomicAdd(&s_beta[l], sig * w_2[dout]);   // ds_add_f32
  }
  __syncthreads();
  if (t < DD) {
    float acc = 0.f;
    for (int l = 0; l < LL; ++l)
      acc += (s_beta[l] * maskv[b * LL + l]) * sh[l * DD + t];
    selectv[b * DD + t] = acc;
  }
}

// ---------------- K7: score = select @ emb[1:].T via V_WMMA_F32_16X16X4_F32
__global__ void __launch_bounds__(256)
k_score_wmma(const float* __restrict__ selectv, const float* __restrict__ emb,
             float* __restrict__ out) {
  const int NTILES = (NC + 15) / 16;                 // 2694
  int wave = (blockIdx.x * blockDim.x + threadIdx.x) >> 5;
  int lane = threadIdx.x & 31;
  int total = (BB / 16) * NTILES;
  if (wave >= total) return;                         // wave-uniform
  int tm = wave / NTILES;
  int tn = wave % NTILES;
  int m0 = tm * 16;
  int n0 = tn * 16;
  int mr = m0 + (lane & 15);
  int nr = n0 + (lane & 15);
  if (nr >= NC) nr = NC - 1;                         // clamp loads; stores guarded
  const float* arow = selectv + (size_t)mr * DD;
  const float* brow = emb + (size_t)(nr + 1) * DD;   // emb[1:]
  int koff = (lane >> 4) * 2;
  v8f c = {};
  for (int k = 0; k < DD; k += 4) {
    v2f a, bf;
    a.x  = arow[k + koff];
    a.y  = arow[k + koff + 1];
    bf.x = brow[k + koff];
    bf.y = brow[k + koff + 1];
    c = __builtin_amdgcn_wmma_f32_16x16x4_f32(false, a, false, bf,
                                              (short)0, c, false, false);
  }
  int nOut = n0 + (lane & 15);
  if (nOut < NC) {
    int mbase = m0 + ((lane >> 4) << 3);
#pragma unroll
    for (int v = 0; v < 8; ++v)
      out[(size_t)(mbase + v) * NC + nOut] = c[v];
  }
}

// ---------------- K8: row softmax over 43097 classes ----------------
__global__ void __launch_bounds__(256)
k_softmax(float* __restrict__ out) {
  int b = blockIdx.x, t = threadIdx.x;
  float* row = out + (size_t)b * NC;
  __shared__ float red[256];
  float mx = -3.4e38f;
  for (int n = t; n < NC; n += 256) mx = fmaxf(mx, row[n]);
  red[t] = mx; __syncthreads();
  for (int s = 128; s > 0; s >>= 1) {
    if (t < s) red[t] = fmaxf(red[t], red[t + s]);
    __syncthreads();
  }
  mx = red[0]; __syncthreads();
  float sum = 0.f;
  for (int n = t; n < NC; n += 256) sum += expf(row[n] - mx);
  red[t] = sum; __syncthreads();
  for (int s = 128; s > 0; s >>= 1) {
    if (t < s) red[t] += red[t + s];
    __syncthreads();
  }
  float inv = 1.f / red[0];
  for (int n = t; n < NC; n += 256) row[n] = expf(row[n] - mx) * inv;
}

// ---------------- launcher ----------------
extern "C" void kernel_launch(void* const* d_in, const int* in_sizes, int n_in,
                              void* d_out, int out_size, void* d_ws, size_t ws_size,
                              hipStream_t stream) {
  const int*   alias_inputs = (const int*)  d_in[0];
  const int*   items        = (const int*)  d_in[1];
  const int*   adj          = (const int*)  d_in[2];
  const float* mask_item    = (const float*)d_in[3];
  const int*   seq_features = (const int*)  d_in[4];
  const int*   adj_all      = (const int*)  d_in[5];
  const float* num_w        = (const float*)d_in[6];
  const float* emb          = (const float*)d_in[7];
  const float* pos_emb      = (const float*)d_in[8];
  const float* a0           = (const float*)d_in[9];
  const float* a1           = (const float*)d_in[10];
  const float* a2           = (const float*)d_in[11];
  const float* a3           = (const float*)d_in[12];
  const float* g_w1         = (const float*)d_in[13];
  const float* g_w2         = (const float*)d_in[14];
  const float* g_w3         = (const float*)d_in[15];
  const float* w_1          = (const float*)d_in[16];
  const float* w_2          = (const float*)d_in[17];
  const float* glu1_w       = (const float*)d_in[18];
  const float* glu1_b       = (const float*)d_in[19];
  const float* glu2_w       = (const float*)d_in[20];
  const float* glu2_b       = (const float*)d_in[21];
  float* outv = (float*)d_out;

  // workspace layout (floats): ~26.5 MB total
  float* ws         = (float*)d_ws;
  float* h          = ws;
  float* h_local    = h          + (size_t)BB * LL * DD;
  float* sess       = h_local    + (size_t)BB * LL * DD;
  float* h_combine  = sess       + (size_t)BB * DD;
  float* seq_hidden = h_combine  + (size_t)BB * LL * DD;
  float* hs         = seq_hidden + (size_t)BB * LL * DD;
  float* selectv    = hs         + (size_t)BB * DD;

  k_h_gather<<<(BB * LL * DD + 255) / 256, 256, 0, stream>>>(items, emb, h);
  k_local<<<BB, 256, 0, stream>>>(adj, a0, a1, a2, a3, h, h_local);
  k_sess<<<BB, 128, 0, stream>>>(seq_features, mask_item, emb, sess);
  k_global<<<BB * LL / 4, 128, 0, stream>>>(items, adj_all, num_w, emb, sess,
                                            g_w1, g_w2, g_w3, h, h_local,
                                            h_combine);
  k_seqhid<<<BB, 128, 0, stream>>>(alias_inputs, mask_item, h_combine,
                                   seq_hidden, hs);
  k_select<<<BB, 256, 0, stream>>>(mask_item, pos_emb, w_1, w_2,
                                   glu1_w, glu1_b, glu2_w, glu2_b,
                                   seq_hidden, hs, selectv);
  int ntiles = (NC + 15) / 16;
  int waves  = (BB / 16) * ntiles;
  int blocks = (waves + 7) / 8;   // 8 waves per 256-thread block
  k_score_wmma<<<blocks, 256, 0, stream>>>(selectv, emb, outv);
  k_softmax<<<BB, 256, 0, stream>>>(outv);
}